// Tacotron2Loss_17360257810668
// MI455X (gfx1250) — compile-verified
//
#include <hip/hip_runtime.h>
#include <math.h>

// ---- CDNA5 wave32 types ----
typedef __attribute__((ext_vector_type(2))) float v2f;
typedef __attribute__((ext_vector_type(8))) float v8f;

#define NB_MSE  1024
#define NB_GATE 64
#define N_DEC   1000
#define T_ENC   250
#define BATCH   64
#define NMEL4   1280000   // 64*80*1000 / 4
#define NMEL_F  5120000.0f
#define GATE4   16000     // 64*1000 / 4
#define GATE_F  64000.0f
#define NSPK    128
#define NAUG    10
#define ZTOT    1024      // 64*16

// Full-wave (32 lane) sum. Uses V_WMMA_F32_16X16X4_F32 with A = ones(16x4),
// B VGPR0 = B VGPR1 = v. Every column n of B then contains exactly
// {v(n), v(n), v(n+16), v(n+16)} across the 4 K-rows (regardless of which
// K maps to which VGPR/half-wave), so D[m,n] = 2*(v(n)+v(n+16)) for all m.
// c[0]*0.5 gives the pairwise half-wave sum per column; 4 xor-shuffles
// finish the 16 columns. EXEC must be all-1s at the call site.
__device__ __forceinline__ float wave_sum(float v) {
  v2f a; a[0] = 1.0f; a[1] = 1.0f;
  v2f b; b[0] = v;    b[1] = v;
  v8f c = {};
  c = __builtin_amdgcn_wmma_f32_16x16x4_f32(false, a, false, b,
                                            (short)0, c, false, false);
  float d = c[0] * 0.5f;              // v(lane%16) + v(lane%16 + 16)
  d += __shfl_xor(d, 1, 32);
  d += __shfl_xor(d, 2, 32);
  d += __shfl_xor(d, 4, 32);
  d += __shfl_xor(d, 8, 32);
  return d;                            // full 32-lane sum in every lane
}

// Block sum; result valid on threadIdx.x == 0 only. Call with full block
// (all lanes of every wave active).
__device__ __forceinline__ float block_sum(float v, float* red) {
  const int lane = threadIdx.x & 31;
  const int wid  = threadIdx.x >> 5;
  float w = wave_sum(v);
  __syncthreads();                     // protect red across repeated calls
  if (lane == 0) red[wid] = w;
  __syncthreads();
  float r = 0.0f;
  if (threadIdx.x == 0) {
    const int nw = blockDim.x >> 5;
    for (int i = 0; i < nw; ++i) r += red[i];
  }
  return r;
}

__device__ __forceinline__ float bce_term(float x, float y) {
  // max(x,0) - x*y + log1p(exp(-|x|))
  return fmaxf(x, 0.0f) - x * y + log1pf(expf(-fabsf(x)));
}

// ---------------- Kernel 1: mel MSE partials (61.4 MB stream) -------------
__global__ void k_mse_partials(const float4* __restrict__ mo,
                               const float4* __restrict__ mp,
                               const float4* __restrict__ mt,
                               float* __restrict__ out2) {
  __shared__ float red[8];
  float s0 = 0.0f, s1 = 0.0f;
  const int stride = gridDim.x * blockDim.x;
  for (int i = blockIdx.x * blockDim.x + threadIdx.x; i < NMEL4; i += stride) {
    float4 a = mo[i], p = mp[i], t = mt[i];
    float d;
    d = a.x - t.x; s0 += d * d;  d = a.y - t.y; s0 += d * d;
    d = a.z - t.z; s0 += d * d;  d = a.w - t.w; s0 += d * d;
    d = p.x - t.x; s1 += d * d;  d = p.y - t.y; s1 += d * d;
    d = p.z - t.z; s1 += d * d;  d = p.w - t.w; s1 += d * d;
  }
  float b0 = block_sum(s0, red);
  float b1 = block_sum(s1, red);
  if (threadIdx.x == 0) {
    out2[2 * blockIdx.x + 0] = b0;
    out2[2 * blockIdx.x + 1] = b1;
  }
}

// ---------------- Kernel 2: gate BCE partials (512 KB stream) -------------
__global__ void k_gate_partials(const float4* __restrict__ x,
                                const float4* __restrict__ y,
                                float* __restrict__ out1) {
  __shared__ float red[8];
  float s = 0.0f;
  const int stride = gridDim.x * blockDim.x;
  for (int i = blockIdx.x * blockDim.x + threadIdx.x; i < GATE4; i += stride) {
    float4 xv = x[i], yv = y[i];
    s += bce_term(xv.x, yv.x) + bce_term(xv.y, yv.y)
       + bce_term(xv.z, yv.z) + bce_term(xv.w, yv.w);
  }
  float b = block_sum(s, red);
  if (threadIdx.x == 0) out1[blockIdx.x] = b;
}

// -------- Kernel 3: guided-attention partials (64 MB stream) --------------
// One block per n-row: compute g[n,t] once (250 exps/block, 250K total
// instead of 16M), then stream the 64 batch rows at that n.
__global__ void k_align_partials(const float* __restrict__ al,
                                 float* __restrict__ out1) {
  __shared__ float red[8];
  const int n = blockIdx.x;            // grid = N_DEC
  const int t = threadIdx.x;
  float acc = 0.0f;
  if (t < T_ENC) {
    float d = (float)n * (1.0f / (float)N_DEC) - (float)t * (1.0f / (float)T_ENC);
    float g = 1.0f - expf(-d * d * (1.0f / 0.4f));   // GA_K = 0.4
    const float* base = al + (long)n * T_ENC + t;
    #pragma unroll 4
    for (int b = 0; b < BATCH; ++b)
      acc += base[(long)b * (N_DEC * T_ENC)] * g;
  }
  // guard closed; EXEC full again for the WMMA reduction
  float s = block_sum(acc, red);
  if (threadIdx.x == 0) out1[n] = s;
}

// -------- Kernel 4: final combine (tiny; 1 block) -------------------------
__global__ void k_final(const float* __restrict__ ws,
                        const float* __restrict__ r_mu,
                        const float* __restrict__ r_lv,
                        const float* __restrict__ spk,
                        const float* __restrict__ aug,
                        const float* __restrict__ spk_t,
                        const float* __restrict__ aug_t,
                        const int*   __restrict__ step,
                        float* __restrict__ out) {
  __shared__ float red[8];
  const int tid = threadIdx.x;

  // mel partial reduce
  float s0 = 0.0f, s1 = 0.0f;
  for (int i = tid; i < NB_MSE; i += blockDim.x) {
    s0 += ws[2 * i + 0];
    s1 += ws[2 * i + 1];
  }
  float S0 = block_sum(s0, red);
  float S1 = block_sum(s1, red);

  // gate partial reduce
  float s2 = (tid < NB_GATE) ? ws[2 * NB_MSE + tid] : 0.0f;
  float S2 = block_sum(s2, red);

  // align partial reduce
  float s3 = 0.0f;
  for (int i = tid; i < N_DEC; i += blockDim.x)
    s3 += ws[2 * NB_MSE + NB_GATE + i];
  float S3 = block_sum(s3, red);

  // KL sum over (64,16)
  float sk = 0.0f;
  for (int i = tid; i < ZTOT; i += blockDim.x) {
    float mu = r_mu[i], lv = r_lv[i];
    sk += 1.0f + lv - mu * mu - expf(lv);
  }
  float SK = block_sum(sk, red);

  // speaker cross-entropy: one thread per batch row
  float sp = 0.0f;
  if (tid < BATCH) {
    const float* tt = spk_t + tid * NSPK;
    int lab = 0; float bm = tt[0];
    for (int j = 1; j < NSPK; ++j) { float v = tt[j]; if (v > bm) { bm = v; lab = j; } }
    const float* xx = spk + tid * NSPK;
    float m = xx[0];
    for (int j = 1; j < NSPK; ++j) m = fmaxf(m, xx[j]);
    float se = 0.0f;
    for (int j = 0; j < NSPK; ++j) se += expf(xx[j] - m);
    sp = (m + logf(se)) - xx[lab];
  }
  float SSP = block_sum(sp, red);

  // augmentation cross-entropy
  float au = 0.0f;
  if (tid < BATCH) {
    const float* tt = aug_t + tid * NAUG;
    int lab = 0; float bm = tt[0];
    for (int j = 1; j < NAUG; ++j) { float v = tt[j]; if (v > bm) { bm = v; lab = j; } }
    const float* xx = aug + tid * NAUG;
    float m = xx[0];
    for (int j = 1; j < NAUG; ++j) m = fmaxf(m, xx[j]);
    float se = 0.0f;
    for (int j = 0; j < NAUG; ++j) se += expf(xx[j] - m);
    au = (m + logf(se)) - xx[lab];
  }
  float SAU = block_sum(au, red);

  if (tid == 0) {
    float mel_loss   = S0 * (1.0f / NMEL_F) + S1 * (1.0f / NMEL_F);
    float gate_loss  = S2 * (1.0f / GATE_F);
    float align_loss = fabsf(S3);
    float rkl        = -0.5f * SK;
    float w  = 1.0f / (1.0f + expf(-0.0025f * ((float)step[0] - 10000.0f)));
    float spl = SSP * (1.0f / (float)BATCH);
    float aul = SAU * (1.0f / (float)BATCH);
    float tmp = mel_loss + gate_loss;
    float total = 10.0f * tmp + 0.01f * (rkl * w)
                + 0.1f * spl + 0.1f * aul + 0.0005f * align_loss;
    out[0] = total;
    out[1] = tmp;
    out[2] = rkl;
    out[3] = w;
    out[4] = spl;
    out[5] = aul;
    out[6] = align_loss;
  }
}

extern "C" void kernel_launch(void* const* d_in, const int* in_sizes, int n_in,
                              void* d_out, int out_size, void* d_ws, size_t ws_size,
                              hipStream_t stream) {
  (void)in_sizes; (void)n_in; (void)out_size; (void)ws_size;
  float* ws = (float*)d_ws;

  // inputs (setup_inputs order):
  // 0 mel_out, 1 mel_out_postnet, 2 gate_out, 3 alignment, 4 r_mu, 5 r_lv,
  // 6 speaker_out, 7 aug_out, 8 mel_target, 9 gate_target,
  // 10 speaker_target, 11 aug_target, 12 step
  k_mse_partials<<<NB_MSE, 256, 0, stream>>>(
      (const float4*)d_in[0], (const float4*)d_in[1], (const float4*)d_in[8], ws);

  k_gate_partials<<<NB_GATE, 256, 0, stream>>>(
      (const float4*)d_in[2], (const float4*)d_in[9], ws + 2 * NB_MSE);

  k_align_partials<<<N_DEC, 256, 0, stream>>>(
      (const float*)d_in[3], ws + 2 * NB_MSE + NB_GATE);

  k_final<<<1, 256, 0, stream>>>(
      ws,
      (const float*)d_in[4], (const float*)d_in[5],
      (const float*)d_in[6], (const float*)d_in[7],
      (const float*)d_in[10], (const float*)d_in[11],
      (const int*)d_in[12],
      (float*)d_out);
}